// Attention_1494648619656
// MI455X (gfx1250) — compile-verified
//
#include <hip/hip_runtime.h>

// ---------------------------------------------------------------------------
// Self-attention block (SAGAN style) for MI455X / gfx1250, fp32 WMMA path.
//   f = Wf x ; g = Wg x ; h = Wh x            (1x1 convs == channel GEMMs)
//   scores[c] = f[c] @ g[c]                   (per-channel 256x256x256 GEMM)
//   P = softmax(scores, axis=1)               (softmax over row index i)
//   out = x + h[c] @ P[c]
// GEMMs use V_WMMA_F32_16X16X4_F32 (fp32 in, fp32 accumulate) with the B tile
// staged in LDS once per workgroup (shared by all 8 waves), fed to WMMA via
// ds_load_b64. LDS fill uses GLOBAL_LOAD_ASYNC_TO_LDS when available.
// ---------------------------------------------------------------------------

#define C_DIM 512
#define N_DIM 256
#define S_DIM (N_DIM * N_DIM)   // 65536
#define KC    32                // K-chunk staged in LDS per barrier pair

typedef float v2f __attribute__((ext_vector_type(2)));
typedef float v8f __attribute__((ext_vector_type(8)));

#if defined(__has_builtin)
#  if __has_builtin(__builtin_amdgcn_global_load_async_to_lds_b32) && \
      __has_builtin(__builtin_amdgcn_s_wait_asynccnt)
#    define USE_ASYNC_LDS 1
#  endif
#endif
#ifndef USE_ASYNC_LDS
#  define USE_ASYNC_LDS 0
#endif

#if USE_ASYNC_LDS
typedef __attribute__((address_space(1))) int gas_int;   // global (AS1)
typedef __attribute__((address_space(3))) int las_int;   // LDS (AS3)
#endif

__device__ __forceinline__ v8f wmma_f32(v2f a, v2f b, v8f c) {
    // 8 args: (neg_a, A, neg_b, B, c_mod, C, reuse_a, reuse_b)
    return __builtin_amdgcn_wmma_f32_16x16x4_f32(
        false, a, false, b, (short)0, c, false, false);
}

// Stage a KC x 64 B-tile (rows src[row0+k][col0..col0+63]) into LDS,
// pair-interleaved so that a (k, k+1) element pair for one column is one
// 8-byte ds_load: lds[(k>>1)*128 + col*2 + (k&1)].
// 256 threads: thread t handles col = t&63, rows k = (t>>6) + 4*p.
__device__ __forceinline__ void stage_btile(
    float* __restrict__ btile, const float* __restrict__ src,
    int row0, long row_stride, int col0, int tid)
{
    const int ls  = tid & 63;
    const int lk0 = tid >> 6;
#if USE_ASYNC_LDS
    #pragma unroll
    for (int p = 0; p < KC / 4; ++p) {
        const int k = lk0 + 4 * p;
        const float* gp = src + (long)(row0 + k) * row_stride + (col0 + ls);
        float* lp = btile + ((k >> 1) * 128 + ls * 2 + (k & 1));
        __builtin_amdgcn_global_load_async_to_lds_b32(
            (gas_int*)gp, (las_int*)lp, 0, 0);
    }
    __builtin_amdgcn_s_wait_asynccnt(0);
#else
    #pragma unroll
    for (int p = 0; p < KC / 4; ++p) {
        const int k = lk0 + 4 * p;
        btile[(k >> 1) * 128 + ls * 2 + (k & 1)] =
            src[(long)(row0 + k) * row_stride + (col0 + ls)];
    }
#endif
}

// ---------------------------------------------------------------------------
// Kernel 1: f,g,h = W{f,g,h} @ x   ([C x C] * [C x S])
// Block = 256 threads = 8 waves; wave w -> channel tile m0=(bx*8+w)*16,
// spatial tile s0 = by*64. 12 v8f accumulators per wave (f,g,h x 4 N-tiles).
// x tile (B) staged in LDS once per workgroup per K-chunk.
// ---------------------------------------------------------------------------
__global__ __launch_bounds__(256) void fgh_kernel(
    const float* __restrict__ x,
    const float* __restrict__ Wf,
    const float* __restrict__ Wg,
    const float* __restrict__ Wh,
    float* __restrict__ f,
    float* __restrict__ g,
    float* __restrict__ h)
{
    __shared__ float btile[KC * 64];   // 8 KB

    const int tid  = threadIdx.x;
    const int lane = tid & 31;
    const int wave = tid >> 5;
    const int half = lane >> 4;
    const int ln16 = lane & 15;

    const int m0 = (blockIdx.x * 8 + wave) * 16;
    const int s0 = blockIdx.y * 64;

    v8f accf[4] = {{0.f,0.f,0.f,0.f,0.f,0.f,0.f,0.f}};
    v8f accg[4] = {{0.f,0.f,0.f,0.f,0.f,0.f,0.f,0.f}};
    v8f acch[4] = {{0.f,0.f,0.f,0.f,0.f,0.f,0.f,0.f}};
    #pragma unroll
    for (int t = 1; t < 4; ++t) { accf[t] = accf[0]; accg[t] = accg[0]; acch[t] = acch[0]; }

    for (int kc = 0; kc < C_DIM; kc += KC) {
        __syncthreads();
        stage_btile(btile, x, kc, S_DIM, s0, tid);
        __syncthreads();

        #pragma unroll 2
        for (int kk = 0; kk < KC; kk += 4) {
            const int ka    = kk + 2 * half;   // even
            const int kglob = kc + ka;
            const v2f af = *(const v2f*)(Wf + (size_t)(m0 + ln16) * C_DIM + kglob);
            const v2f ag = *(const v2f*)(Wg + (size_t)(m0 + ln16) * C_DIM + kglob);
            const v2f ah = *(const v2f*)(Wh + (size_t)(m0 + ln16) * C_DIM + kglob);
            #pragma unroll
            for (int t = 0; t < 4; ++t) {
                const v2f b = *(const v2f*)(btile + ((ka >> 1) * 128 + (t * 16 + ln16) * 2));
                accf[t] = wmma_f32(af, b, accf[t]);
                accg[t] = wmma_f32(ag, b, accg[t]);
                acch[t] = wmma_f32(ah, b, acch[t]);
            }
        }
    }

    // C/D layout: VGPR r, lanes 0-15 -> M = r, lanes 16-31 -> M = r + 8
    #pragma unroll
    for (int t = 0; t < 4; ++t) {
        #pragma unroll
        for (int r = 0; r < 8; ++r) {
            const int c = m0 + r + 8 * half;
            const size_t idx = (size_t)c * S_DIM + (s0 + t * 16 + ln16);
            f[idx] = accf[t][r];
            g[idx] = accg[t][r];
            h[idx] = acch[t][r];
        }
    }
}

// ---------------------------------------------------------------------------
// Kernel 2: per-channel GEMM  scores[c] = f[c] @ g[c]   (256x256, K=256)
// grid = (2, 4, 512): 8 waves cover M in [i0, i0+128), each wave 16x64.
// g tile (B) staged in LDS (shared by all 8 waves).
// ---------------------------------------------------------------------------
__global__ __launch_bounds__(256) void scores_kernel(
    const float* __restrict__ f,
    const float* __restrict__ g,
    float* __restrict__ scores)
{
    __shared__ float btile[KC * 64];

    const int tid  = threadIdx.x;
    const int lane = tid & 31;
    const int wave = tid >> 5;
    const int half = lane >> 4;
    const int ln16 = lane & 15;

    const int c  = blockIdx.z;
    const float* fc = f + (size_t)c * S_DIM;
    const float* gc = g + (size_t)c * S_DIM;
    float*       sc = scores + (size_t)c * S_DIM;

    const int i0 = (blockIdx.x * 8 + wave) * 16;
    const int j0 = blockIdx.y * 64;

    v8f acc[4] = {{0.f,0.f,0.f,0.f,0.f,0.f,0.f,0.f}};
    #pragma unroll
    for (int t = 1; t < 4; ++t) acc[t] = acc[0];

    for (int kc = 0; kc < N_DIM; kc += KC) {
        __syncthreads();
        stage_btile(btile, gc, kc, N_DIM, j0, tid);
        __syncthreads();

        #pragma unroll 2
        for (int kk = 0; kk < KC; kk += 4) {
            const int ka = kk + 2 * half;
            const v2f a = *(const v2f*)(fc + (size_t)(i0 + ln16) * N_DIM + (kc + ka));
            #pragma unroll
            for (int t = 0; t < 4; ++t) {
                const v2f b = *(const v2f*)(btile + ((ka >> 1) * 128 + (t * 16 + ln16) * 2));
                acc[t] = wmma_f32(a, b, acc[t]);
            }
        }
    }

    #pragma unroll
    for (int t = 0; t < 4; ++t) {
        #pragma unroll
        for (int r = 0; r < 8; ++r) {
            sc[(size_t)(i0 + r + 8 * half) * N_DIM + (j0 + t * 16 + ln16)] = acc[t][r];
        }
    }
}

// ---------------------------------------------------------------------------
// Kernel 3: softmax over axis=1 (row index i) for each (c, j) column, in place.
// One thread per column; consecutive threads -> consecutive j -> coalesced.
// ---------------------------------------------------------------------------
__global__ __launch_bounds__(256) void softmax_kernel(float* __restrict__ scores)
{
    const int gid = blockIdx.x * 256 + threadIdx.x;   // 512*256 threads total
    const int c = gid >> 8;
    const int j = gid & (N_DIM - 1);
    float* base = scores + (size_t)c * S_DIM + j;

    float m = -3.0e38f;
    float s = 0.0f;
    #pragma unroll 4
    for (int i = 0; i < N_DIM; ++i) {
        const float v = base[(size_t)i * N_DIM];
        const float mn = fmaxf(m, v);
        s = s * __expf(m - mn) + __expf(v - mn);
        m = mn;
    }
    const float inv = 1.0f / s;
    #pragma unroll 4
    for (int i = 0; i < N_DIM; ++i) {
        const float v = base[(size_t)i * N_DIM];
        base[(size_t)i * N_DIM] = __expf(v - m) * inv;
    }
}

// ---------------------------------------------------------------------------
// Kernel 4: out[c] = x[c] + h[c] @ P[c]   (same tiling as scores_kernel,
// residual add fused into the epilogue)
// ---------------------------------------------------------------------------
__global__ __launch_bounds__(256) void attn_out_kernel(
    const float* __restrict__ h,
    const float* __restrict__ P,
    const float* __restrict__ x,
    float* __restrict__ out)
{
    __shared__ float btile[KC * 64];

    const int tid  = threadIdx.x;
    const int lane = tid & 31;
    const int wave = tid >> 5;
    const int half = lane >> 4;
    const int ln16 = lane & 15;

    const int c  = blockIdx.z;
    const float* hc = h + (size_t)c * S_DIM;
    const float* pc = P + (size_t)c * S_DIM;
    const float* xc = x + (size_t)c * S_DIM;
    float*       oc = out + (size_t)c * S_DIM;

    const int i0 = (blockIdx.x * 8 + wave) * 16;
    const int j0 = blockIdx.y * 64;

    v8f acc[4] = {{0.f,0.f,0.f,0.f,0.f,0.f,0.f,0.f}};
    #pragma unroll
    for (int t = 1; t < 4; ++t) acc[t] = acc[0];

    for (int kc = 0; kc < N_DIM; kc += KC) {
        __syncthreads();
        stage_btile(btile, pc, kc, N_DIM, j0, tid);
        __syncthreads();

        #pragma unroll 2
        for (int kk = 0; kk < KC; kk += 4) {
            const int ka = kk + 2 * half;
            const v2f a = *(const v2f*)(hc + (size_t)(i0 + ln16) * N_DIM + (kc + ka));
            #pragma unroll
            for (int t = 0; t < 4; ++t) {
                const v2f b = *(const v2f*)(btile + ((ka >> 1) * 128 + (t * 16 + ln16) * 2));
                acc[t] = wmma_f32(a, b, acc[t]);
            }
        }
    }

    #pragma unroll
    for (int t = 0; t < 4; ++t) {
        #pragma unroll
        for (int r = 0; r < 8; ++r) {
            const size_t idx = (size_t)(i0 + r + 8 * half) * N_DIM + (j0 + t * 16 + ln16);
            oc[idx] = xc[idx] + acc[t][r];
        }
    }
}

// ---------------------------------------------------------------------------
// Host-side launcher. Workspace layout (floats): f | g | h | scores
// = 4 * C*S * 4 bytes = 512 MiB.
// ---------------------------------------------------------------------------
extern "C" void kernel_launch(void* const* d_in, const int* in_sizes, int n_in,
                              void* d_out, int out_size, void* d_ws, size_t ws_size,
                              hipStream_t stream) {
    const float* x  = (const float*)d_in[0];
    const float* Wf = (const float*)d_in[1];
    const float* Wg = (const float*)d_in[2];
    const float* Wh = (const float*)d_in[3];
    float* out = (float*)d_out;

    const size_t plane = (size_t)C_DIM * S_DIM;   // 33,554,432 floats
    float* f      = (float*)d_ws;
    float* g      = f + plane;
    float* h      = g + plane;
    float* scores = h + plane;

    // 1) f,g,h = W{f,g,h} @ x
    fgh_kernel<<<dim3(C_DIM / 128, S_DIM / 64), 256, 0, stream>>>(
        x, Wf, Wg, Wh, f, g, h);

    // 2) scores[c] = f[c] @ g[c]
    scores_kernel<<<dim3(N_DIM / 128, N_DIM / 64, C_DIM), 256, 0, stream>>>(
        f, g, scores);

    // 3) softmax over axis=1, in place
    softmax_kernel<<<dim3((C_DIM * N_DIM) / 256), 256, 0, stream>>>(scores);

    // 4) out = x + h[c] @ P[c]
    attn_out_kernel<<<dim3(N_DIM / 128, N_DIM / 64, C_DIM), 256, 0, stream>>>(
        h, scores, x, out);
}